// BahdanauAttention_76982993814149
// MI455X (gfx1250) — compile-verified
//
#include <hip/hip_runtime.h>
#include <hip/hip_bf16.h>

// Problem constants (match reference)
static constexpr int B_ = 32;
static constexpr int S_ = 2048;
static constexpr int H_ = 1024;
static constexpr int V_ = 2 * H_;   // value feature dim = 2048

typedef __attribute__((ext_vector_type(2))) float v2f;
typedef __attribute__((ext_vector_type(8))) float v8f;
typedef __attribute__((ext_vector_type(2))) float fvec2;
typedef __attribute__((ext_vector_type(4))) float fvec4;

// GCC-style int4 vector: matches the async-to-LDS builtin's pointee type
// ('__attribute__((__vector_size__(4 * sizeof(int)))) int') exactly.
typedef int i4vec __attribute__((vector_size(16)));
typedef __attribute__((address_space(1))) i4vec* glb_i4_p;  // printed: __device__
typedef __attribute__((address_space(3))) i4vec* lds_i4_p;  // printed: __shared__

// ---------------------------------------------------------------------------
// Fast tanh: gfx1250 has native v_tanh_f32 (single TRANS op). Fallback uses
// exp2 + rcp (2 TRANS ops): tanh(x) = 1 - 2 / (exp2(2*log2e*x) + 1)
// ---------------------------------------------------------------------------
__device__ __forceinline__ float fast_tanh(float x) {
#if __has_builtin(__builtin_amdgcn_tanhf)
    return __builtin_amdgcn_tanhf(x);
#else
    float e = __builtin_amdgcn_exp2f(x * 2.8853900817779268f); // 2*log2(e)
    return 1.0f - 2.0f * __builtin_amdgcn_rcpf(e + 1.0f);
#endif
}

// ---------------------------------------------------------------------------
// Kernel 1: proj_query[b,n] = sum_h query[b,h] * Wq[n,h] + bq[n]
// M=32 (2 tiles), N=1024 (64 tiles), K=1024.  One wave (32 threads) per
// 16x16 output tile, V_WMMA_F32_16X16X4_F32 over K in steps of 4.
// ---------------------------------------------------------------------------
__global__ __launch_bounds__(32) void k_proj_query(
    const float* __restrict__ query, const float* __restrict__ Wq,
    const float* __restrict__ bq, float* __restrict__ pq) {
  const int mt   = blockIdx.x & 1;    // 2 tiles over B
  const int nt   = blockIdx.x >> 1;   // 64 tiles over H
  const int lane = threadIdx.x;
  const int half = lane >> 4;
  const int l    = lane & 15;
  const int m    = mt * 16 + l;       // A row this lane supplies
  const int n    = nt * 16 + l;       // B col this lane supplies

#if __has_builtin(__builtin_amdgcn_wmma_f32_16x16x4_f32)
  v8f c = {};
  for (int k = 0; k < H_; k += 4) {
    v2f a, b;
    const float* qrow = query + m * H_ + k + half * 2;
    a.x = qrow[0];
    a.y = qrow[1];
    // B[K][n] = Wq[n*H + K]
    const float* wrow = Wq + n * H_ + k + half * 2;
    b.x = wrow[0];
    b.y = wrow[1];
    c = __builtin_amdgcn_wmma_f32_16x16x4_f32(
        /*neg_a=*/false, a, /*neg_b=*/false, b,
        /*c_mod=*/(short)0, c, /*reuse_a=*/false, /*reuse_b=*/false);
  }
  const float bias = bq[n];
#pragma unroll
  for (int r = 0; r < 8; ++r) {
    const int m_out = mt * 16 + r + 8 * half;
    pq[m_out * H_ + n] = c[r] + bias;
  }
#else
  // Scalar fallback producing identical outputs (same tile ownership).
  const float bias = bq[n];
  for (int r = 0; r < 8; ++r) {
    const int m_out = mt * 16 + r + 8 * half;
    float acc = 0.f;
    for (int k = 0; k < H_; ++k)
      acc = fmaf(query[m_out * H_ + k], Wq[n * H_ + k], acc);
    pq[m_out * H_ + n] = acc + bias;
  }
#endif
}

// ---------------------------------------------------------------------------
// Kernel 2: energy[b,s] = sum_h tanh(pq[b,h] + proj_key[b,s,h]) * v_energy[h]
// 256 threads = 8 waves per block; each wave owns one s. Each batch owns
// exactly 256 consecutive blocks, so the whole block shares one pq row:
// stage pq[b,:] and v_energy into LDS (async-to-LDS DMA when available),
// then stream proj_key with non-temporal 128-bit loads (stream-once data,
// keep it out of L2). Wave32 butterfly reduction at the end.
// ---------------------------------------------------------------------------
__global__ __launch_bounds__(256) void k_energy(
    const float* __restrict__ proj_key, const float* __restrict__ pq,
    const float* __restrict__ v_energy, float* __restrict__ energy) {
  __shared__ float s_pq[H_];
  __shared__ float s_v[H_];
  const int b   = blockIdx.x >> 8;          // 256 blocks per batch (S/8)
  const int tid = threadIdx.x;

#if __has_builtin(__builtin_amdgcn_global_load_async_to_lds_b128) && \
    __has_builtin(__builtin_amdgcn_s_wait_asynccnt)
  // CDNA5 async global->LDS copy (ASYNCcnt-tracked): each thread moves one
  // 16-byte chunk of each array; 256 threads x 16B = 4KB = one full H row.
  {
    const int i = tid * 4;
    __builtin_amdgcn_global_load_async_to_lds_b128(
        (glb_i4_p)(pq + b * H_ + i), (lds_i4_p)&s_pq[i], 0, 0);
    __builtin_amdgcn_global_load_async_to_lds_b128(
        (glb_i4_p)(v_energy + i), (lds_i4_p)&s_v[i], 0, 0);
    __builtin_amdgcn_s_wait_asynccnt(0);
  }
#else
  for (int i = tid; i < H_; i += 256) {
    s_pq[i] = pq[b * H_ + i];
    s_v[i]  = v_energy[i];
  }
#endif
  __syncthreads();

  const int wave = tid >> 5, lane = tid & 31;
  const int s = (blockIdx.x & 255) * 8 + wave;
  const float* pk = proj_key + ((size_t)b * S_ + s) * H_;

  float sum = 0.f;
#pragma unroll
  for (int i = 0; i < 8; ++i) {
    const int h = i * 128 + lane * 4;
    const fvec4 p = __builtin_nontemporal_load((const fvec4*)(pk + h));
    sum += fast_tanh(p.x + s_pq[h + 0]) * s_v[h + 0];
    sum += fast_tanh(p.y + s_pq[h + 1]) * s_v[h + 1];
    sum += fast_tanh(p.z + s_pq[h + 2]) * s_v[h + 2];
    sum += fast_tanh(p.w + s_pq[h + 3]) * s_v[h + 3];
  }
#pragma unroll
  for (int off = 16; off; off >>= 1) sum += __shfl_xor(sum, off, 32);
  if (lane == 0) energy[b * S_ + s] = sum;
}

// ---------------------------------------------------------------------------
// Kernel 3: alphas[b,s] = softmax over s of masked energy. One block per b.
// ---------------------------------------------------------------------------
__global__ __launch_bounds__(256) void k_softmax(
    const float* __restrict__ energy, const int* __restrict__ mask,
    float* __restrict__ alphas) {
  __shared__ float red[8];
  const int b   = blockIdx.x;
  const int tid = threadIdx.x;
  const float NEG_INF = -__builtin_inff();
  const float LOG2E   = 1.4426950408889634f;

  float mx = NEG_INF;
  for (int s = tid; s < S_; s += 256) {
    const float e = mask[b * S_ + s] ? energy[b * S_ + s] : NEG_INF;
    mx = fmaxf(mx, e);
  }
#pragma unroll
  for (int off = 16; off; off >>= 1) mx = fmaxf(mx, __shfl_xor(mx, off, 32));
  if ((tid & 31) == 0) red[tid >> 5] = mx;
  __syncthreads();
  mx = red[0];
#pragma unroll
  for (int i = 1; i < 8; ++i) mx = fmaxf(mx, red[i]);
  __syncthreads();

  float sum = 0.f;
  for (int s = tid; s < S_; s += 256) {
    const float e = mask[b * S_ + s] ? energy[b * S_ + s] : NEG_INF;
    sum += __builtin_amdgcn_exp2f((e - mx) * LOG2E);
  }
#pragma unroll
  for (int off = 16; off; off >>= 1) sum += __shfl_xor(sum, off, 32);
  if ((tid & 31) == 0) red[tid >> 5] = sum;
  __syncthreads();
  sum = 0.f;
#pragma unroll
  for (int i = 0; i < 8; ++i) sum += red[i];
  const float inv = 1.0f / sum;

  for (int s = tid; s < S_; s += 256) {
    const float e = mask[b * S_ + s] ? energy[b * S_ + s] : NEG_INF;
    alphas[b * S_ + s] = __builtin_amdgcn_exp2f((e - mx) * LOG2E) * inv;
  }
}

// ---------------------------------------------------------------------------
// Kernel 4: context[b,v] = sum_s energy_masked[b,s] * value[b,s,v]
// (reference uses PRE-softmax masked energy here, not alphas).
// Grid: (4 col-chunks, 32 batches) x 256 threads; each thread owns TWO
// columns (64-bit per-lane loads). value is a 512MB stream-once read ->
// non-temporal hint keeps it out of L2. 4-way s-unroll = 4 outstanding
// b64 loads per thread for memory-level parallelism. energy/mask reads are
// wave-uniform (scalar path). Fully deterministic (no atomics).
// ---------------------------------------------------------------------------
__global__ __launch_bounds__(256) void k_context(
    const float* __restrict__ energy, const int* __restrict__ mask,
    const float* __restrict__ value, float* __restrict__ context) {
  const int b  = blockIdx.y;
  const int v2 = blockIdx.x * 256 + threadIdx.x;   // float2-column index 0..1023
  const float NEG_INF = -__builtin_inff();

  const fvec2* val = (const fvec2*)(value + (size_t)b * S_ * V_) + v2;
  const float* er  = energy + b * S_;
  const int*   mr  = mask + b * S_;

  float ax0 = 0.f, ay0 = 0.f, ax1 = 0.f, ay1 = 0.f;
  float ax2 = 0.f, ay2 = 0.f, ax3 = 0.f, ay3 = 0.f;
  for (int s = 0; s < S_; s += 4) {
    const float e0 = mr[s + 0] ? er[s + 0] : NEG_INF;
    const float e1 = mr[s + 1] ? er[s + 1] : NEG_INF;
    const float e2 = mr[s + 2] ? er[s + 2] : NEG_INF;
    const float e3 = mr[s + 3] ? er[s + 3] : NEG_INF;
    const fvec2 w0 = __builtin_nontemporal_load(val + (size_t)(s + 0) * (V_ / 2));
    const fvec2 w1 = __builtin_nontemporal_load(val + (size_t)(s + 1) * (V_ / 2));
    const fvec2 w2 = __builtin_nontemporal_load(val + (size_t)(s + 2) * (V_ / 2));
    const fvec2 w3 = __builtin_nontemporal_load(val + (size_t)(s + 3) * (V_ / 2));
    ax0 = fmaf(e0, w0.x, ax0); ay0 = fmaf(e0, w0.y, ay0);
    ax1 = fmaf(e1, w1.x, ax1); ay1 = fmaf(e1, w1.y, ay1);
    ax2 = fmaf(e2, w2.x, ax2); ay2 = fmaf(e2, w2.y, ay2);
    ax3 = fmaf(e3, w3.x, ax3); ay3 = fmaf(e3, w3.y, ay3);
  }
  fvec2 outv;
  outv.x = (ax0 + ax1) + (ax2 + ax3);
  outv.y = (ay0 + ay1) + (ay2 + ay3);
  *(fvec2*)(context + b * V_ + v2 * 2) = outv;
}

// ---------------------------------------------------------------------------
extern "C" void kernel_launch(void* const* d_in, const int* in_sizes, int n_in,
                              void* d_out, int out_size, void* d_ws,
                              size_t ws_size, hipStream_t stream) {
  (void)in_sizes; (void)n_in; (void)out_size; (void)ws_size;
  const float* query    = (const float*)d_in[0];  // (B,1,H)
  const float* proj_key = (const float*)d_in[1];  // (B,S,H)
  const float* value    = (const float*)d_in[2];  // (B,S,2H)
  const int*   src_mask = (const int*)d_in[3];    // (B,1,S)
  const float* Wq       = (const float*)d_in[4];  // (H,H)
  const float* bq       = (const float*)d_in[5];  // (H,)
  const float* v_energy = (const float*)d_in[6];  // (H,)

  float* out     = (float*)d_out;
  float* context = out;               // B*2H = 65536 floats
  float* alphas  = out + B_ * V_;     // B*S  = 65536 floats

  float* pq     = (float*)d_ws;       // B*H  = 32768 floats
  float* energy = pq + B_ * H_;       // B*S  = 65536 floats

  // 1) projection GEMM (WMMA f32 16x16x4): 2 x 64 tiles, one wave each
  k_proj_query<<<dim3(2 * (H_ / 16)), dim3(32), 0, stream>>>(query, Wq, bq, pq);
  // 2) energy: 8 waves/block, one s per wave
  k_energy<<<dim3(B_ * S_ / 8), dim3(256), 0, stream>>>(proj_key, pq, v_energy,
                                                        energy);
  // 3) softmax -> alphas
  k_softmax<<<dim3(B_), dim3(256), 0, stream>>>(energy, src_mask, alphas);
  // 4) context = energy @ value (float2 per thread, NT stream)
  k_context<<<dim3(V_ / 512, B_), dim3(256), 0, stream>>>(energy, src_mask,
                                                          value, context);
}